// InteractionBlock_13889924235382
// MI455X (gfx1250) — compile-verified
//
#include <hip/hip_runtime.h>

#define FEAT   128
#define EVD    15
#define D      131          // FEAT + 3 invariants
#define KPAD   132          // D padded to multiple of 4 (row stride of Wl / cat)
#define NTILES 9            // ceil(131 / 16) output column tiles
#define CPAD   144          // 9 * 16 padded output columns (rows of Wl)
#define NODES_PER_BLOCK 128
#define THREADS 256

typedef __attribute__((ext_vector_type(2))) float v2f;
typedef __attribute__((ext_vector_type(4))) float v4f;
typedef __attribute__((ext_vector_type(8))) float v8f;
typedef __attribute__((ext_vector_type(4))) int   v4i;

#if defined(__has_builtin)
#if __has_builtin(__builtin_amdgcn_global_load_async_to_lds_b128)
#define HAVE_ASYNC_B128 1
#endif
#if __has_builtin(__builtin_amdgcn_s_wait_asynccnt)
#define HAVE_WAIT_ASYNC 1
#endif
#endif

#if defined(HAVE_ASYNC_B128)
typedef __attribute__((address_space(1))) v4i gv4i;   // global 16B vector
typedef __attribute__((address_space(3))) v4i lv4i;   // LDS 16B vector
#define GLOBAL_V4(p) ((gv4i*)(uintptr_t)(p))
#define LDS_V4(p)    ((lv4i*)(uint32_t)(uintptr_t)(p))
#endif

static __device__ __forceinline__ void wait_async_zero() {
#if defined(HAVE_ASYNC_B128)
#if defined(HAVE_WAIT_ASYNC)
    __builtin_amdgcn_s_wait_asynccnt(0);
#else
    asm volatile("s_wait_asynccnt 0" ::: "memory");
#endif
#endif
}

__global__ __launch_bounds__(THREADS)
void interaction_block_kernel(const float* __restrict__ inv,
                              const float* __restrict__ ev,
                              const float* __restrict__ W,
                              const float* __restrict__ bias,
                              float* __restrict__ out_inv,
                              float* __restrict__ out_ev,
                              int n_nodes)
{
    // Wl[j][k] = W[j][k], rows padded to 132 (8B-aligned b64 fragments),
    // extra zero rows j = 131..143 for the last column tile.
    __shared__ float Wl[CPAD * KPAD];                 // 76,032 B
    __shared__ float cat[NODES_PER_BLOCK * KPAD];     // 67,584 B

    const int tid  = threadIdx.x;
    const int lane = tid & 31;
    const int wave = tid >> 5;
    const long long nodeBase = (long long)blockIdx.x * NODES_PER_BLOCK;

    // ---- Zero-fill padding of Wl: column k=131 of rows j<131, and rows 131..143.
    //      Disjoint from the copy region -> no extra barrier needed.
    for (int idx = tid; idx < D + 13 * KPAD; idx += THREADS) {
        int a = (idx < D) ? (idx * KPAD + (KPAD - 1)) : (D * KPAD + (idx - D));
        Wl[a] = 0.0f;
    }

    // ---- Copy W (131x131) into Wl with b128 global reads ----
    {
        const int NQ4 = (D * D) >> 2;                 // 4290 full 16B chunks (+1 tail elt)
        for (int q4 = tid; q4 < NQ4; q4 += THREADS) {
            int q = q4 << 2;
            v4f w = *(const v4f*)(W + q);             // hipMalloc base: 16B aligned
            float vals[4] = {w.x, w.y, w.z, w.w};
            #pragma unroll
            for (int u = 0; u < 4; ++u) {
                int qq = q + u;
                int j  = qq / D;                      // magic-multiply division
                int k  = qq - j * D;
                Wl[j * KPAD + k] = vals[u];
            }
        }
        if (tid == 0) Wl[130 * KPAD + 130] = W[D * D - 1];  // tail element
    }

    // ---- Stage inv features into cat rows (b128; async->LDS when available) ----
    for (int idx = tid; idx < NODES_PER_BLOCK * (FEAT / 4); idx += THREADS) {
        int m  = idx >> 5;                            // node-in-block (32 x 16B per row)
        int c4 = idx & 31;
        long long node = nodeBase + m;
        if (node >= n_nodes) node = n_nodes - 1;      // clamp; stores guarded later
        const float* g = inv + node * FEAT + 4 * c4;
        float*       l = cat + m * KPAD + 4 * c4;     // KPAD%4==0 -> 16B aligned
#if defined(HAVE_ASYNC_B128)
        __builtin_amdgcn_global_load_async_to_lds_b128(GLOBAL_V4(g), LDS_V4(l), 0, 0);
#else
        v4f v = __builtin_nontemporal_load((const v4f*)g);
        *(v4f*)l = v;
#endif
    }

    // ---- ev invariants (segment sums of ev*ev) into cat cols 128..131 ----
    if (tid < NODES_PER_BLOCK) {
        int m = tid;
        long long node = nodeBase + m;
        if (node >= n_nodes) node = n_nodes - 1;
        const float* e = ev + node * EVD;
        float s1 = 0.f, s2 = 0.f, s3 = 0.f;
        #pragma unroll
        for (int q = 0; q < 3;  ++q) s1 += e[q] * e[q];
        #pragma unroll
        for (int q = 3; q < 8;  ++q) s2 += e[q] * e[q];
        #pragma unroll
        for (int q = 8; q < 15; ++q) s3 += e[q] * e[q];
        cat[m * KPAD + 128] = s1;
        cat[m * KPAD + 129] = s2;
        cat[m * KPAD + 130] = s3;
        cat[m * KPAD + 131] = 0.0f;                   // K padding
    }

    wait_async_zero();      // my async->LDS writes landed
    __syncthreads();        // everyone's staging visible

    // ---- WMMA main loop: 16 nodes / wave, 9 column tiles, K = 132 in steps of 4 ----
    const int m0  = wave * 16;
    const int lhi = lane >> 4;     // 0: lanes 0-15, 1: lanes 16-31
    const int llo = lane & 15;

    v8f acc[NTILES];
    #pragma unroll
    for (int t = 0; t < NTILES; ++t) acc[t] = (v8f){0,0,0,0,0,0,0,0};

    const v2f* catp = (const v2f*)cat;   // row stride 66 v2f
    const v2f* Wlp  = (const v2f*)Wl;    // row stride 66 v2f
    const int  arow = (m0 + llo) * (KPAD / 2);
    const int  brow = llo * (KPAD / 2);

    for (int s = 0; s < KPAD / 4; ++s) {
        // A fragment (16x4 f32): lane half selects K pair (0,1) vs (2,3)
        v2f a = catp[arow + 2 * s + lhi];
        #pragma unroll
        for (int t = 0; t < NTILES; ++t) {
            // B fragment (4x16 f32): (W[j][k], W[j][k+1]) contiguous in Wl row j
            v2f b = Wlp[brow + t * 16 * (KPAD / 2) + 2 * s + lhi];
            acc[t] = __builtin_amdgcn_wmma_f32_16x16x4_f32(
                false, a, false, b, (short)0, acc[t], false, false);
        }
    }

    // ---- Epilogue phase 1: bias add; cols<128 transposed into this wave's own
    //      cat rows (wave-private, DS in-order per wave); cols 128..130 -> d_ev.
    #pragma unroll
    for (int t = 0; t < NTILES; ++t) {
        int col = t * 16 + llo;
        if (col >= D) continue;
        float bv = bias[col];
        #pragma unroll
        for (int r = 0; r < 8; ++r) {
            int m = m0 + r + 8 * lhi;                 // C/D layout: M = r + 8*laneHi
            float val = acc[t][r] + bv;
            if (col < FEAT) {
                cat[m * KPAD + col] = val;            // LDS transpose staging
            } else {
                long long node = nodeBase + m;
                if (node < n_nodes) {
                    int seg = col - FEAT;             // 0,1,2
                    int off = (seg == 0) ? 0 : (seg == 1 ? 3 : 8);
                    int len = 2 * seg + 3;            // 3,5,7
                    const float* e = ev + node * EVD;
                    float* o = out_ev + node * EVD;
                    for (int q = 0; q < len; ++q)
                        __builtin_nontemporal_store(val * e[off + q], o + off + q);
                }
            }
        }
    }

    // ---- Epilogue phase 2: stream each node row out as wave-wide b128 stores ----
    asm volatile("s_wait_dscnt 0" ::: "memory");      // my LDS writes -> my reads
    #pragma unroll
    for (int m = 0; m < 16; ++m) {
        long long node = nodeBase + m0 + m;
        if (node < n_nodes) {
            v4f vv = *(const v4f*)(cat + (m0 + m) * KPAD + 4 * lane);
            __builtin_nontemporal_store(vv, (v4f*)(out_inv + node * FEAT + 4 * lane));
        }
    }
}

extern "C" void kernel_launch(void* const* d_in, const int* in_sizes, int n_in,
                              void* d_out, int out_size, void* d_ws, size_t ws_size,
                              hipStream_t stream) {
    const float* inv  = (const float*)d_in[0];   // [N, 128]
    const float* ev   = (const float*)d_in[1];   // [N, 15]
    const float* W    = (const float*)d_in[2];   // [131, 131]
    const float* bias = (const float*)d_in[3];   // [131]

    const int n_nodes = in_sizes[0] / FEAT;
    float* out_inv = (float*)d_out;                                  // [N,128]
    float* out_ev  = (float*)d_out + (long long)n_nodes * FEAT;      // [N,15]

    const int grid = (n_nodes + NODES_PER_BLOCK - 1) / NODES_PER_BLOCK;
    hipLaunchKernelGGL(interaction_block_kernel, dim3(grid), dim3(THREADS), 0, stream,
                       inv, ev, W, bias, out_inv, out_ev, n_nodes);
}